// SMI_56118042689958
// MI455X (gfx1250) — compile-verified
//
#include <hip/hip_runtime.h>

// ---------------- problem constants ----------------
#define Bv  4
#define Lv  48
#define Tv  16
#define NWv 20
#define Dv  256
#define DLv 64
#define Rv  (Bv*Lv*Lv*Tv)      // 147456 rows of f_c
#define Gv  (Bv*Lv*Lv)         // 9216 (b,l,m) groups (16 rows each)
#define BLLD (Bv*Lv*Lv*Dv)     // 2359296
#define BLD  (Bv*Lv*Dv)        // 49152

typedef float v2f __attribute__((ext_vector_type(2)));
typedef float v8f __attribute__((ext_vector_type(8)));

// ---------------- CDNA5 async memory->LDS staging ----------------
#if defined(__has_builtin)
# if __has_builtin(__builtin_amdgcn_global_load_async_to_lds_b128)
#  define USE_ASYNC_LDS 1
# endif
#endif
#ifndef USE_ASYNC_LDS
# define USE_ASYNC_LDS 0
#endif

#if USE_ASYNC_LDS
// builtin signature (from compiler diagnostic): param1 is
//   'int __attribute__((vector_size(16))) __device__ *'  (global, AS1)
// param2 is the LDS-side pointer (AS3); args 3/4 are imm offset / cpol.
typedef int v4i __attribute__((vector_size(16)));
typedef v4i __attribute__((address_space(1))) gv4i;
typedef v4i __attribute__((address_space(3))) lv4i;
// copy 16 bytes global -> LDS, tracked by ASYNCcnt
__device__ __forceinline__ void async_cp16(const float* gsrc, float* ldst) {
  __builtin_amdgcn_global_load_async_to_lds_b128((gv4i*)gsrc, (lv4i*)ldst, 0, 0);
}
__device__ __forceinline__ void wait_async0() {
# if __has_builtin(__builtin_amdgcn_s_wait_asynccnt)
  __builtin_amdgcn_s_wait_asynccnt(0);
# else
  asm volatile("s_wait_asynccnt 0x0" ::: "memory");
# endif
}
#endif

// V_WMMA_F32_16X16X4_F32 : D(16x16,f32) = A(16x4,f32) * B(4x16,f32) + C
__device__ __forceinline__ v8f wmma4(v2f a, v2f b, v8f c) {
  return __builtin_amdgcn_wmma_f32_16x16x4_f32(false, a, false, b, (short)0, c,
                                               false, false);
}

// =====================================================================
// gmfm[b,l,m,d] = sigmoid(mu * f_s[b,d]) * mu
// =====================================================================
__global__ void k_gmfm(const float* __restrict__ mu, const float* __restrict__ fs,
                       float* __restrict__ out, int total) {
  int i = blockIdx.x * 256 + threadIdx.x;
  if (i >= total) return;
  int d = i & (Dv - 1);
  int b = i / (Lv * Lv * Dv);
  float m = mu[i];
  float x = m * fs[b * Dv + d];
  float sg = 1.0f / (1.0f + __expf(-x));
  out[i] = sg * m;
}

// =====================================================================
// Generic small projection: Y[r,j] = X[r,:K] . W[j,:K] + bias[j]
// =====================================================================
__global__ void k_proj(const float* __restrict__ X, const float* __restrict__ W,
                       const float* __restrict__ bias, float* __restrict__ Y,
                       int rows, int K, int N) {
  int i = blockIdx.x * 256 + threadIdx.x;
  if (i >= rows * N) return;
  int r = i / N, j = i - r * N;
  const float* xr = X + (size_t)r * K;
  const float* wr = W + (size_t)j * K;
  float acc = bias[j];
  for (int k = 0; k < K; ++k) acc = fmaf(xr[k], wr[k], acc);
  Y[i] = acc;
}

// =====================================================================
// Down-projection GEMM (WMMA): Fh[r, j] = cu[r,:256] . dcc_w[j,:256] + b[j]
// 1 block = 16 rows x 64 cols, 4 waves (one 16x16 tile each), K = 256.
// A tile staged via GLOBAL_LOAD_ASYNC_TO_LDS_B128 (ASYNCcnt path).
// =====================================================================
__global__ __launch_bounds__(128) void k_gemm_down(
    const float* __restrict__ A, const float* __restrict__ W,
    const float* __restrict__ bias, float* __restrict__ Y) {
  __shared__ __attribute__((aligned(16))) float As[16 * 260];  // padded stride
  const int r0 = blockIdx.x * 16;
  const int tid = threadIdx.x;
#if USE_ASYNC_LDS
  for (int i = 0; i < 8; ++i) {              // 1024 x 16B chunks / 128 thr
    int c = tid + 128 * i;                   // chunk id
    int row = c >> 6, col = (c & 63) * 4;
    async_cp16(A + (size_t)(r0 + row) * Dv + col, As + row * 260 + col);
  }
  wait_async0();
#else
  for (int i = 0; i < 32; ++i) {
    int flat = tid + 128 * i;
    int row = flat >> 8, col = flat & 255;
    As[row * 260 + col] = A[(size_t)(r0 + row) * Dv + col];
  }
#endif
  __syncthreads();
  const int wave = tid >> 5, lane = tid & 31;
  const int l15 = lane & 15, half = lane >> 4;
  const int n0 = wave * 16;
  v8f c = {};
  const float* arow = As + l15 * 260 + 2 * half;
  const float* wrow = W + (size_t)(n0 + l15) * Dv + 2 * half;
  for (int k0 = 0; k0 < Dv; k0 += 4) {
    v2f a = *(const v2f*)(arow + k0);
    v2f b = *(const v2f*)(wrow + k0);
    c = wmma4(a, b, c);
  }
  const int col = n0 + l15;
  const float bv = bias[col];
  const int rowbase = half * 8;
  for (int i = 0; i < 8; ++i)
    Y[(size_t)(r0 + rowbase + i) * DLv + col] = c[i] + bv;
}

// =====================================================================
// Fused content-unit attention, one block per (b,l,m) group (16 rows, DL=64)
// =====================================================================
__global__ __launch_bounds__(256) void k_group(
    const float* __restrict__ Fh, const float* __restrict__ cWq,
    const float* __restrict__ cbq, const float* __restrict__ khat,
    const float* __restrict__ fwhat, const float* __restrict__ fshat,
    float* __restrict__ Fcc) {
  __shared__ __attribute__((aligned(16))) float Fh_s[16 * 64];
  __shared__ __attribute__((aligned(16))) float q_s[16 * 64];
  __shared__ __attribute__((aligned(16))) float Fq_s[16 * 68];  // padded
  __shared__ __attribute__((aligned(16))) float sc[16 * NWv];
  __shared__ __attribute__((aligned(16))) float G_s[16 * 20];   // padded
  const int g = blockIdx.x;
  const int b = g / (Lv * Lv);
  const int tid = threadIdx.x;

#if USE_ASYNC_LDS
  async_cp16(Fh + (size_t)g * 16 * DLv + tid * 4, Fh_s + tid * 4);  // 256x16B
  wait_async0();
#else
  for (int i = 0; i < 4; ++i) {
    int flat = tid + 256 * i;
    Fh_s[flat] = Fh[(size_t)g * 16 * DLv + flat];
  }
#endif
  __syncthreads();

  for (int i = 0; i < 4; ++i) {                    // q = Fh @ cWq^T + cbq
    int flat = tid + 256 * i;
    int row = flat >> 6, j = flat & 63;
    float acc = cbq[j];
    const float* fr = Fh_s + row * 64;
    const float* wr = cWq + j * DLv;
    for (int k = 0; k < DLv; ++k) acc = fmaf(fr[k], wr[k], acc);
    q_s[flat] = acc;
  }
  __syncthreads();

  for (int idx = tid; idx < 16 * NWv; idx += 256) {  // scores q . k_hat / 8
    int t = idx / NWv, n = idx - t * NWv;
    const float* qr = q_s + t * 64;
    const float* kr = khat + (size_t)(b * NWv + n) * DLv;
    float s = 0.f;
    for (int k = 0; k < DLv; ++k) s = fmaf(qr[k], kr[k], s);
    sc[t * NWv + n] = s * 0.125f;
  }
  __syncthreads();

  if (tid < 16) {                                  // softmax over 20 words
    float* r = sc + tid * NWv;
    float mx = r[0];
    for (int n = 1; n < NWv; ++n) mx = fmaxf(mx, r[n]);
    float sum = 0.f;
    for (int n = 0; n < NWv; ++n) { float e = __expf(r[n] - mx); r[n] = e; sum += e; }
    float inv = 1.f / sum;
    for (int n = 0; n < NWv; ++n) r[n] *= inv;
  }
  __syncthreads();

  for (int i = 0; i < 4; ++i) {                    // f_cq
    int flat = tid + 256 * i;
    int row = flat >> 6, j = flat & 63;
    float fca = 0.f;
    const float* ar = sc + row * NWv;
    for (int n = 0; n < NWv; ++n)
      fca = fmaf(ar[n], fwhat[(size_t)(b * NWv + n) * DLv + j], fca);
    Fq_s[row * 68 + j] = Fh_s[row * 64 + j] * (fca + fshat[b * DLv + j]);
  }
  __syncthreads();

  if (tid < 32) {                                  // Gram via WMMA (A == B)
    const int l15 = tid & 15, half = tid >> 4;
    v8f c = {};
    const float* fr = Fq_s + l15 * 68 + 2 * half;
    for (int k0 = 0; k0 < DLv; k0 += 4) {
      v2f a = *(const v2f*)(fr + k0);
      c = wmma4(a, a, c);
    }
    const int rowbase = half * 8;
    for (int i = 0; i < 8; ++i) G_s[(rowbase + i) * 20 + l15] = c[i] * 0.125f;
  }
  __syncthreads();

  if (tid < 16) {                                  // softmax over s (16)
    float* r = G_s + tid * 20;
    float mx = r[0];
    for (int s = 1; s < 16; ++s) mx = fmaxf(mx, r[s]);
    float sum = 0.f;
    for (int s = 0; s < 16; ++s) { float e = __expf(r[s] - mx); r[s] = e; sum += e; }
    float inv = 1.f / sum;
    for (int s = 0; s < 16; ++s) r[s] *= inv;
  }
  __syncthreads();

  const int wave = tid >> 5;
  if (wave < 4) {                                  // Fcc = A_c @ Fh (WMMA, K=16)
    const int lane = tid & 31, l15 = lane & 15, half = lane >> 4;
    const int n0 = wave * 16;
    v8f c = {};
    for (int k0 = 0; k0 < 16; k0 += 4) {
      v2f a = *(const v2f*)(G_s + l15 * 20 + k0 + 2 * half);
      v2f bb;
      bb.x = Fh_s[(k0 + 2 * half) * 64 + n0 + l15];
      bb.y = Fh_s[(k0 + 2 * half + 1) * 64 + n0 + l15];
      c = wmma4(a, bb, c);
    }
    const int rowbase = half * 8;
    for (int i = 0; i < 8; ++i)
      Fcc[(size_t)(g * 16 + rowbase + i) * DLv + n0 + l15] = c[i];
  }
}

// =====================================================================
// Up-projection GEMM (WMMA) + fused epilogue (bias + residual + gate)
// and fused T-mean.  1 block = 16 rows x 256 cols = 16 waves.
// =====================================================================
__global__ __launch_bounds__(512) void k_gemm_up(
    const float* __restrict__ Fcc, const float* __restrict__ W,
    const float* __restrict__ bias, const float* __restrict__ gmfm,
    float* __restrict__ cu, float* __restrict__ fcmean) {
  __shared__ __attribute__((aligned(16))) float As[16 * 68];
  __shared__ __attribute__((aligned(16))) float msum[512];
  const int g = blockIdx.x;
  const int r0 = g * 16;
  const int tid = threadIdx.x;
#if USE_ASYNC_LDS
  if (tid < 256) {                           // 256 x 16B chunks
    int row = tid >> 4, col = (tid & 15) * 4;
    async_cp16(Fcc + (size_t)(r0 + row) * DLv + col, As + row * 68 + col);
    wait_async0();
  }
#else
  for (int i = 0; i < 2; ++i) {
    int flat = tid + 512 * i;
    int row = flat >> 6, col = flat & 63;
    As[row * 68 + col] = Fcc[(size_t)(r0 + row) * DLv + col];
  }
#endif
  __syncthreads();
  const int wave = tid >> 5, lane = tid & 31;
  const int l15 = lane & 15, half = lane >> 4;
  const int n0 = wave * 16;
  v8f c = {};
  const float* arow = As + l15 * 68 + 2 * half;
  const float* wrow = W + (size_t)(n0 + l15) * DLv + 2 * half;
  for (int k0 = 0; k0 < DLv; k0 += 4) {
    v2f a = *(const v2f*)(arow + k0);
    v2f b = *(const v2f*)(wrow + k0);
    c = wmma4(a, b, c);
  }
  const int col = n0 + l15;
  const int rowbase = half * 8;
  const float bv = bias[col];
  const float ga = gmfm[(size_t)g * Dv + col];
  float s = 0.f;
  for (int i = 0; i < 8; ++i) {
    size_t idx = (size_t)(r0 + rowbase + i) * Dv + col;
    float v = c[i] + bv + cu[idx] + ga;
    cu[idx] = v;
    s += v;
  }
  msum[tid] = s;
  __syncthreads();
  if (lane < 16) {  // combine the two 8-row halves of this column
    float tot = msum[wave * 32 + lane] + msum[wave * 32 + lane + 16];
    fcmean[(size_t)g * Dv + n0 + lane] = tot * (1.0f / 16.0f);
  }
}

// =====================================================================
// Boundary unit, stage 1
// =====================================================================
__global__ __launch_bounds__(256) void k_bq(
    const float* __restrict__ qb, const float* __restrict__ kb,
    const float* __restrict__ fw, const float* __restrict__ fs,
    const float* __restrict__ bu, float* __restrict__ fbq) {
  __shared__ float sc[NWv];
  const int bl = blockIdx.x;
  const int b = bl / Lv;
  const int tid = threadIdx.x;
  if (tid < NWv) {
    const float* qr = qb + (size_t)bl * Dv;
    const float* kr = kb + (size_t)(b * NWv + tid) * Dv;
    float s = 0.f;
    for (int d = 0; d < Dv; ++d) s = fmaf(qr[d], kr[d], s);
    sc[tid] = s * (1.0f / 16.0f);
  }
  __syncthreads();
  if (tid == 0) {
    float mx = sc[0];
    for (int n = 1; n < NWv; ++n) mx = fmaxf(mx, sc[n]);
    float sum = 0.f;
    for (int n = 0; n < NWv; ++n) { float e = __expf(sc[n] - mx); sc[n] = e; sum += e; }
    float inv = 1.f / sum;
    for (int n = 0; n < NWv; ++n) sc[n] *= inv;
  }
  __syncthreads();
  const int d = tid;
  float fba = 0.f;
  for (int n = 0; n < NWv; ++n)
    fba = fmaf(sc[n], fw[(size_t)(b * NWv + n) * Dv + d], fba);
  size_t idx = (size_t)bl * Dv + d;
  fbq[idx] = bu[idx] * (fba + fs[b * Dv + d]);
}

// =====================================================================
// Boundary unit, stage 2
// =====================================================================
__global__ __launch_bounds__(256) void k_bupdate(
    const float* __restrict__ fbq, const float* __restrict__ bu,
    const float* __restrict__ gmfm, float* __restrict__ bu_new) {
  __shared__ float aw[Lv];
  const int bl = blockIdx.x;
  const int b = bl / Lv;
  const int tid = threadIdx.x;
  if (tid < Lv) {
    const float* x = fbq + (size_t)bl * Dv;
    const float* y = fbq + (size_t)(b * Lv + tid) * Dv;
    float s = 0.f;
    for (int d = 0; d < Dv; ++d) s = fmaf(x[d], y[d], s);
    aw[tid] = s * (1.0f / 16.0f);
  }
  __syncthreads();
  if (tid == 0) {
    float mx = aw[0];
    for (int n = 1; n < Lv; ++n) mx = fmaxf(mx, aw[n]);
    float sum = 0.f;
    for (int n = 0; n < Lv; ++n) { float e = __expf(aw[n] - mx); aw[n] = e; sum += e; }
    float inv = 1.f / sum;
    for (int n = 0; n < Lv; ++n) aw[n] *= inv;
  }
  __syncthreads();
  const int d = tid;
  float fbb = 0.f, fbm = 0.f;
  for (int m = 0; m < Lv; ++m) {
    float a = aw[m];
    fbb = fmaf(a, bu[(size_t)(b * Lv + m) * Dv + d], fbb);
    fbm = fmaf(a, gmfm[((size_t)bl * Lv + m) * Dv + d], fbm);
  }
  size_t idx = (size_t)bl * Dv + d;
  bu_new[idx] = fbb + bu[idx] + fbm;
}

// =====================================================================
// Moment unit
// =====================================================================
__global__ void k_wb(const float* __restrict__ mfb_w,
                     const float* __restrict__ bu_new, float* __restrict__ wb) {
  int i = blockIdx.x * 256 + threadIdx.x;  // over B*L*D
  if (i >= BLD) return;
  int d = i & (Dv - 1);
  int bo = i >> 8;
  int b = bo / Lv, o = bo - b * Lv;
  float acc = 0.f;
  for (int k = 0; k < Lv; ++k)
    acc = fmaf(mfb_w[o * Lv + k], bu_new[(size_t)(b * Lv + k) * Dv + d], acc);
  wb[i] = acc;
}

__global__ void k_moment(const float* __restrict__ wb,
                         const float* __restrict__ bu_new,
                         const float* __restrict__ mfb_b,
                         const float* __restrict__ mfc_w,
                         const float* __restrict__ mfc_b,
                         const float* __restrict__ fcmean,
                         const float* __restrict__ mu, float* __restrict__ mu_new,
                         int total) {
  int i = blockIdx.x * 256 + threadIdx.x;  // over B*L*L*D
  if (i >= total) return;
  int d = i & (Dv - 1);
  int rem = i >> 8;
  int m = rem % Lv;
  int bo = rem / Lv;
  int b = bo / Lv, o = bo - b * Lv;
  float acc = wb[(size_t)bo * Dv + d] * bu_new[(size_t)(b * Lv + m) * Dv + d] +
              mfb_b[o] + mfc_b[o];
  const float* wr = mfc_w + o * Lv;
  for (int k = 0; k < Lv; ++k)
    acc = fmaf(wr[k], fcmean[((size_t)(b * Lv + k) * Lv + m) * Dv + d], acc);
  mu_new[i] = acc + mu[i];
}

// =====================================================================
// Host side
// =====================================================================
extern "C" void kernel_launch(void* const* d_in, const int* in_sizes, int n_in,
                              void* d_out, int out_size, void* d_ws, size_t ws_size,
                              hipStream_t stream) {
  (void)in_sizes; (void)n_in; (void)out_size;
  const float* f_c   = (const float*)d_in[0];
  const float* f_m   = (const float*)d_in[1];
  const float* f_b   = (const float*)d_in[2];
  const float* f_w   = (const float*)d_in[3];
  const float* f_s   = (const float*)d_in[4];
  const float* bWq   = (const float*)d_in[5];
  const float* bbq   = (const float*)d_in[6];
  const float* bWk   = (const float*)d_in[7];
  const float* bbk   = (const float*)d_in[8];
  const float* cWq   = (const float*)d_in[9];
  const float* cbq   = (const float*)d_in[10];
  const float* cWk   = (const float*)d_in[11];
  const float* cbk   = (const float*)d_in[12];
  const float* dcc_w = (const float*)d_in[13];
  const float* dcc_b = (const float*)d_in[14];
  const float* dcw_w = (const float*)d_in[15];
  const float* dcw_b = (const float*)d_in[16];
  const float* dcs_w = (const float*)d_in[17];
  const float* dcs_b = (const float*)d_in[18];
  const float* ucc_w = (const float*)d_in[19];
  const float* ucc_b = (const float*)d_in[20];
  const float* mfb_w = (const float*)d_in[21];
  const float* mfb_b = (const float*)d_in[22];
  const float* mfc_w = (const float*)d_in[23];
  const float* mfc_b = (const float*)d_in[24];

  // ---- workspace layout (floats) ----
  const size_t o_cu = 0;
  const size_t o_Fh     = o_cu     + (size_t)Rv * Dv;      // f_c_hat
  const size_t o_Fcc    = o_Fh     + (size_t)Rv * DLv;     // f_cc_hat
  const size_t o_gmfm   = o_Fcc    + (size_t)Rv * DLv;
  const size_t o_mua    = o_gmfm   + BLLD;
  const size_t o_mub    = o_mua    + BLLD;
  const size_t o_fcmean = o_mub    + BLLD;
  const size_t o_bua    = o_fcmean + BLLD;
  const size_t o_bub    = o_bua    + BLD;
  const size_t o_fwhat  = o_bub    + BLD;
  const size_t o_fshat  = o_fwhat  + Bv * NWv * DLv;
  const size_t o_khat   = o_fshat  + Bv * DLv;
  const size_t o_kb     = o_khat   + Bv * NWv * DLv;
  const size_t o_qb     = o_kb     + Bv * NWv * Dv;
  const size_t o_fbq    = o_qb     + BLD;
  const size_t o_wb     = o_fbq    + BLD;
  const size_t total_f  = o_wb     + BLD;
  if (ws_size < total_f * sizeof(float)) return;  // not enough scratch

  float* ws     = (float*)d_ws;
  float* cu     = ws + o_cu;
  float* Fh     = ws + o_Fh;
  float* Fcc    = ws + o_Fcc;
  float* gmfm   = ws + o_gmfm;
  float* mu_a   = ws + o_mua;
  float* mu_b   = ws + o_mub;
  float* fcmean = ws + o_fcmean;
  float* bu_a   = ws + o_bua;
  float* bu_b   = ws + o_bub;
  float* fwhat  = ws + o_fwhat;
  float* fshat  = ws + o_fshat;
  float* khat   = ws + o_khat;
  float* kb     = ws + o_kb;
  float* qb     = ws + o_qb;
  float* fbq    = ws + o_fbq;
  float* wbuf   = ws + o_wb;

  // state init (inputs must stay unmodified; cu is updated in place)
  (void)hipMemcpyAsync(cu,   f_c, (size_t)Rv * Dv * sizeof(float), hipMemcpyDeviceToDevice, stream);
  (void)hipMemcpyAsync(mu_a, f_m, (size_t)BLLD * sizeof(float),    hipMemcpyDeviceToDevice, stream);
  (void)hipMemcpyAsync(bu_a, f_b, (size_t)BLD * sizeof(float),     hipMemcpyDeviceToDevice, stream);

  // iteration-invariant projections
  k_proj<<<(Bv*NWv*DLv + 255)/256, 256, 0, stream>>>(f_w, dcw_w, dcw_b, fwhat, Bv*NWv, Dv, DLv);
  k_proj<<<(Bv*DLv + 255)/256,     256, 0, stream>>>(f_s, dcs_w, dcs_b, fshat, Bv,     Dv, DLv);
  k_proj<<<(Bv*NWv*DLv + 255)/256, 256, 0, stream>>>(fwhat, cWk, cbk, khat,  Bv*NWv, DLv, DLv);
  k_proj<<<(Bv*NWv*Dv + 255)/256,  256, 0, stream>>>(f_w, bWk, bbk, kb,      Bv*NWv, Dv,  Dv);

  float* mu_cur = mu_a; float* mu_nxt = mu_b;
  float* bu_cur = bu_a; float* bu_nxt = bu_b;

  for (int it = 0; it < 3; ++it) {
    // shared gate tensor  g_m * fm
    k_gmfm<<<(BLLD + 255)/256, 256, 0, stream>>>(mu_cur, f_s, gmfm, BLLD);

    // ---- content unit (WMMA pipeline over f_c) ----
    k_gemm_down<<<Gv, 128, 0, stream>>>(cu, dcc_w, dcc_b, Fh);
    k_group<<<Gv, 256, 0, stream>>>(Fh, cWq, cbq, khat, fwhat, fshat, Fcc);
    k_gemm_up<<<Gv, 512, 0, stream>>>(Fcc, ucc_w, ucc_b, gmfm, cu, fcmean);

    // ---- boundary unit ----
    k_proj<<<(BLD + 255)/256, 256, 0, stream>>>(bu_cur, bWq, bbq, qb, Bv*Lv, Dv, Dv);
    k_bq<<<Bv*Lv, 256, 0, stream>>>(qb, kb, f_w, f_s, bu_cur, fbq);
    k_bupdate<<<Bv*Lv, 256, 0, stream>>>(fbq, bu_cur, gmfm, bu_nxt);

    // ---- moment unit ----
    k_wb<<<(BLD + 255)/256, 256, 0, stream>>>(mfb_w, bu_nxt, wbuf);
    k_moment<<<(BLLD + 255)/256, 256, 0, stream>>>(wbuf, bu_nxt, mfb_b, mfc_w,
                                                   mfc_b, fcmean, mu_cur, mu_nxt, BLLD);

    float* t;
    t = mu_cur; mu_cur = mu_nxt; mu_nxt = t;
    t = bu_cur; bu_cur = bu_nxt; bu_nxt = t;
  }

  float* out = (float*)d_out;  // (mu, bu) concatenated
  (void)hipMemcpyAsync(out,        mu_cur, (size_t)BLLD * sizeof(float), hipMemcpyDeviceToDevice, stream);
  (void)hipMemcpyAsync(out + BLLD, bu_cur, (size_t)BLD * sizeof(float),  hipMemcpyDeviceToDevice, stream);
}